// LocationSensitiveAttention_20005957665081
// MI455X (gfx1250) — compile-verified
//
#include <hip/hip_runtime.h>
#include <hip/hip_bf16.h>

typedef float v2f __attribute__((ext_vector_type(2)));
typedef float v8f __attribute__((ext_vector_type(8)));

#define B_      64
#define T_      2048
#define DQ_     1024
#define DV_     512
#define ATTN_   128
#define FILTERS_ 32
#define KSIZE_  31
#define KPAD_   32   // conv taps padded to 32 (row 31 of CW is zero)

// ---------------------------------------------------------------------------
// Kernel 0: fold conv kernel into CW[k][n] = sum_f ck[k,f] * W_loc[f,n]
//           and fold conv_bias into bias_eff[n] = att_bias[n] + cb@W_loc.
// ---------------------------------------------------------------------------
__global__ void lsa_precompute_kernel(const float* __restrict__ ck,
                                      const float* __restrict__ cb,
                                      const float* __restrict__ wloc,
                                      const float* __restrict__ att_bias,
                                      float* __restrict__ CW,
                                      float* __restrict__ bias_eff) {
    const int n = threadIdx.x;  // 128 threads
    for (int k = 0; k < KPAD_; ++k) {
        float s = 0.f;
        if (k < KSIZE_) {
            for (int f = 0; f < FILTERS_; ++f)
                s += ck[k * FILTERS_ + f] * wloc[f * ATTN_ + n];
        }
        CW[k * ATTN_ + n] = s;
    }
    float b = att_bias[n];
    for (int f = 0; f < FILTERS_; ++f) b += cb[f] * wloc[f * ATTN_ + n];
    bias_eff[n] = b;
}

// ---------------------------------------------------------------------------
// Kernel 1: pq[64][128] = query[64][1024] @ W_q[1024][128], fp32 WMMA 16x16x4.
// One wave per 16x16 output tile; grid = 4 (M tiles) * 8 (N tiles) = 32 waves.
// ---------------------------------------------------------------------------
__global__ __launch_bounds__(32) void lsa_pq_kernel(const float* __restrict__ q,
                                                    const float* __restrict__ wq,
                                                    float* __restrict__ pq) {
    const int lane = threadIdx.x;
    const int half = lane >> 4, lr = lane & 15;
    const int mt = blockIdx.x >> 3;   // 0..3 over B=64
    const int nt = blockIdx.x & 7;    // 0..7 over ATTN=128
    const int m = mt * 16 + lr;
    const int n = nt * 16 + lr;
    v8f c = {};
    for (int k = 0; k < DQ_; k += 4) {
        const int ka = k + 2 * half;          // lanes 0-15: K=k,k+1; 16-31: K=k+2,k+3
        v2f a, b;
        a.x = q[m * DQ_ + ka];
        a.y = q[m * DQ_ + ka + 1];
        b.x = wq[ka * ATTN_ + n];
        b.y = wq[(ka + 1) * ATTN_ + n];
        c = __builtin_amdgcn_wmma_f32_16x16x4_f32(false, a, false, b,
                                                  (short)0, c, false, false);
    }
#pragma unroll
    for (int r = 0; r < 8; ++r) {
        const int row = mt * 16 + r + 8 * half;   // C/D layout: lane<16 -> M=r, lane>=16 -> M=r+8
        pq[row * ATTN_ + n] = c[r];
    }
}

// ---------------------------------------------------------------------------
// Kernel 2: per-batch fused scores + softmax.
// 1 workgroup (512 threads = 16 waves) per batch b.
//   ploc tile (16t x 16n) via WMMA f32 16x16x4 over K=32 (padded conv taps),
//   score e[t] = sum_n v_att[n]*tanh(keys + pq + ploc + bias_eff),
//   softmax over T in LDS, write alignment and attention_state.
// ---------------------------------------------------------------------------
__global__ __launch_bounds__(512) void lsa_score_softmax_kernel(
    const float* __restrict__ keys, const float* __restrict__ state,
    const float* __restrict__ pq, const float* __restrict__ CW,
    const float* __restrict__ bias_eff, const float* __restrict__ v_att,
    float* __restrict__ out_align, float* __restrict__ out_state) {
    __shared__ float sCW[KPAD_ * ATTN_];   // 16 KB
    __shared__ float sE[T_];               // 8 KB scores -> exp
    __shared__ float sPQ[ATTN_], sB[ATTN_], sV[ATTN_];
    __shared__ float red[512];

    const int tid = threadIdx.x;
    const int b = blockIdx.x;

    for (int i = tid; i < KPAD_ * ATTN_; i += 512) sCW[i] = CW[i];
    if (tid < ATTN_) {
        sPQ[tid] = pq[b * ATTN_ + tid];
        sB[tid]  = bias_eff[tid];
        sV[tid]  = v_att[tid];
    }
    __syncthreads();

    const int lane = tid & 31, wv = tid >> 5;   // 16 waves
    const int half = lane >> 4, lr = lane & 15;
    const float* st = state + b * T_;
    const float* ky = keys + (size_t)b * T_ * ATTN_;

    // 128 t-tiles of 16 rows; 8 tiles per wave (uniform -> EXEC all-ones at WMMA)
    for (int tt = wv; tt < T_ / 16; tt += 16) {
        const int t0 = tt * 16;
        const int trow = t0 + lr;              // A-matrix row (M)
        // A fragments: S[t,k] = state[b, t+k-15], zero-padded at edges.
        v2f af[8];
#pragma unroll
        for (int s = 0; s < 8; ++s) {
            const int k = s * 4 + 2 * half;
            const int i0 = trow + k - (KSIZE_ / 2);
            const int i1 = i0 + 1;
            af[s].x = (i0 >= 0 && i0 < T_) ? st[i0] : 0.f;
            af[s].y = (i1 >= 0 && i1 < T_) ? st[i1] : 0.f;
        }
        float acc[8];
#pragma unroll
        for (int r = 0; r < 8; ++r) acc[r] = 0.f;

        for (int j = 0; j < 8; ++j) {          // 8 n-tiles over ATTN=128
            const int n = j * 16 + lr;
            v8f c = {};
#pragma unroll
            for (int s = 0; s < 8; ++s) {
                const int k = s * 4 + 2 * half;
                v2f bf;
                bf.x = sCW[k * ATTN_ + n];
                bf.y = sCW[(k + 1) * ATTN_ + n];
                c = __builtin_amdgcn_wmma_f32_16x16x4_f32(false, af[s], false, bf,
                                                          (short)0, c, false, false);
            }
            const float pb = sPQ[n] + sB[n];
            const float vn = sV[n];
#pragma unroll
            for (int r = 0; r < 8; ++r) {
                const int t = t0 + r + 8 * half;
                const float sc = ky[(size_t)t * ATTN_ + n] + pb + c[r];
                acc[r] += vn * tanhf(sc);
            }
        }
        // Reduce over the 16 lanes of each half (xor 1,2,4,8 stays in-half).
#pragma unroll
        for (int r = 0; r < 8; ++r) {
            float v = acc[r];
            for (int off = 1; off < 16; off <<= 1) v += __shfl_xor(v, off, 32);
            if (lr == 0) sE[t0 + r + 8 * half] = v;
        }
    }
    __syncthreads();

    // Softmax over T in LDS.
    float lm = -3.402823466e38f;
    for (int t = tid; t < T_; t += 512) lm = fmaxf(lm, sE[t]);
    red[tid] = lm; __syncthreads();
    for (int s = 256; s > 0; s >>= 1) {
        if (tid < s) red[tid] = fmaxf(red[tid], red[tid + s]);
        __syncthreads();
    }
    const float m = red[0]; __syncthreads();

    float ls = 0.f;
    for (int t = tid; t < T_; t += 512) {
        const float e = __expf(sE[t] - m);
        sE[t] = e;
        ls += e;
    }
    red[tid] = ls; __syncthreads();
    for (int s = 256; s > 0; s >>= 1) {
        if (tid < s) red[tid] += red[tid + s];
        __syncthreads();
    }
    const float inv = 1.f / red[0];

    for (int t = tid; t < T_; t += 512) {
        const float a = sE[t] * inv;
        out_align[b * T_ + t] = a;
        out_state[b * T_ + t] = a + st[t];
    }
}

// ---------------------------------------------------------------------------
// Kernel 3: attention partials. Grid = B*8 (T split into chunks of 256).
// Thread d in [0,512) accumulates sum_t a[t]*values[b,t,d] over its chunk.
// This streams the 256 MB `values` tensor once, fully coalesced.
// ---------------------------------------------------------------------------
__global__ __launch_bounds__(512) void lsa_attn_partial_kernel(
    const float* __restrict__ values, const float* __restrict__ align,
    float* __restrict__ part) {
    __shared__ float sa[256];
    const int tid = threadIdx.x;          // = d
    const int b = blockIdx.x >> 3;
    const int c = blockIdx.x & 7;
    const int tbeg = c * 256;
    if (tid < 256) sa[tid] = align[b * T_ + tbeg + tid];
    __syncthreads();
    const float* vp = values + ((size_t)b * T_ + tbeg) * DV_ + tid;
    float acc = 0.f;
    for (int t = 0; t < 256; ++t) {
        __builtin_prefetch(vp + (size_t)(t + 16) * DV_, 0, 0);  // global_prefetch_b8 ahead of stream
        acc += sa[t] * vp[(size_t)t * DV_];
    }
    part[((size_t)b * 8 + c) * DV_ + tid] = acc;
}

// ---------------------------------------------------------------------------
// Kernel 4: reduce 8 partials -> attention[b][d].
// ---------------------------------------------------------------------------
__global__ __launch_bounds__(512) void lsa_attn_reduce_kernel(
    const float* __restrict__ part, float* __restrict__ out_att) {
    const int b = blockIdx.x, d = threadIdx.x;
    float acc = 0.f;
#pragma unroll
    for (int c = 0; c < 8; ++c) acc += part[((size_t)b * 8 + c) * DV_ + d];
    out_att[b * DV_ + d] = acc;
}

// ---------------------------------------------------------------------------
extern "C" void kernel_launch(void* const* d_in, const int* in_sizes, int n_in,
                              void* d_out, int out_size, void* d_ws, size_t ws_size,
                              hipStream_t stream) {
    const float* query    = (const float*)d_in[0];
    const float* values   = (const float*)d_in[1];
    const float* keys     = (const float*)d_in[2];
    const float* state    = (const float*)d_in[3];
    const float* W_q      = (const float*)d_in[4];
    const float* ck       = (const float*)d_in[5];
    const float* cb       = (const float*)d_in[6];
    const float* wloc     = (const float*)d_in[7];
    const float* v_att    = (const float*)d_in[8];
    const float* att_bias = (const float*)d_in[9];

    float* out       = (float*)d_out;                // tuple, concatenated flat
    float* out_att   = out;                          // [64, 512]
    float* out_align = out + B_ * DV_;               // [64, 2048]
    float* out_state = out_align + B_ * T_;          // [64, 2048]

    float* ws       = (float*)d_ws;
    float* ws_pq    = ws;                            // 64*128      =   8192 f
    float* ws_CW    = ws + 8192;                     // 32*128      =   4096 f
    float* ws_bias  = ws + 12288;                    // 128 f
    float* ws_part  = ws + 12416;                    // 64*8*512    = 262144 f

    lsa_precompute_kernel<<<1, 128, 0, stream>>>(ck, cb, wloc, att_bias, ws_CW, ws_bias);
    lsa_pq_kernel<<<32, 32, 0, stream>>>(query, W_q, ws_pq);
    lsa_score_softmax_kernel<<<B_, 512, 0, stream>>>(keys, state, ws_pq, ws_CW,
                                                     ws_bias, v_att, out_align, out_state);
    lsa_attn_partial_kernel<<<B_ * 8, 512, 0, stream>>>(values, out_align, ws_part);
    lsa_attn_reduce_kernel<<<B_, 512, 0, stream>>>(ws_part, out_att);
}